// LocalCRF2_11742440587464
// MI455X (gfx1250) — compile-verified
//
#include <hip/hip_runtime.h>

// LocalCRF2 for MI455X (gfx1250, wave32).
// Sizes fixed by the reference: B=4, C=32, H=W=320, NC=21, kernel_len=7.

#define BATCHN 4
#define CCH    32
#define HH     320
#define WW     320
#define NCH    21
#define KL     7
#define PADR   3
#define HWPIX  (HH*WW)          // 102400
#define NPIX   (BATCHN*HWPIX)   // 409600

#define TS   16                 // pixel tile edge
#define HT   (TS + KL - 1)      // 22 halo tile edge
#define HE   (HT*HT)            // 484 halo pixels
#define FSTR 36                 // F LDS pixel stride (32 ch padded to 36, 144B -> 16B aligned)
#define QSTR 24                 // Q LDS pixel stride (21 ch padded to 24, 96B  -> 16B aligned)

#define SIGROWS 32              // padded sigmoid(compat) table: 32 rows x 24 cols
#define SIGCOLS 24

typedef __attribute__((ext_vector_type(2))) float v2f;
typedef __attribute__((ext_vector_type(8))) float v8f;

// ---- optional CDNA5 async global->LDS staging (guarded, cannot break build) -
#if defined(__has_builtin)
#if __has_builtin(__builtin_amdgcn_global_load_async_to_lds_b32) && \
    __has_builtin(__builtin_amdgcn_s_wait_asynccnt)
#define USE_ASYNC_LDS 1
#endif
#endif

#ifdef USE_ASYNC_LDS
typedef __attribute__((address_space(1))) int gbl_i32;   // global (AS1)
typedef __attribute__((address_space(3))) int lds_i32;   // LDS (AS3)
__device__ __forceinline__ void async_copy_b32(const float* g, float* l) {
  __builtin_amdgcn_global_load_async_to_lds_b32((gbl_i32*)g, (lds_i32*)l, 0, 0);
}
__device__ __forceinline__ void async_wait_all() {
  __builtin_amdgcn_s_wait_asynccnt(0);
}
#endif

// ---------------------------------------------------------------- softmax ---
__global__ __launch_bounds__(256) void softmax_nc(const float* __restrict__ logit,
                                                  float* __restrict__ Q) {
  int g = blockIdx.x * 256 + threadIdx.x;
  if (g >= NPIX) return;
  int b = g / HWPIX, p = g % HWPIX;
  const float* src = logit + (size_t)b * NCH * HWPIX + p;
  float v[NCH];
  float m = -3.4e38f;
#pragma unroll
  for (int c = 0; c < NCH; ++c) { v[c] = src[(size_t)c * HWPIX]; m = fmaxf(m, v[c]); }
  float s = 0.f;
#pragma unroll
  for (int c = 0; c < NCH; ++c) { v[c] = __expf(v[c] - m); s += v[c]; }
  float inv = __builtin_amdgcn_rcpf(s);           // v_rcp_f32, no div chain
  float* dst = Q + (size_t)b * NCH * HWPIX + p;
#pragma unroll
  for (int c = 0; c < NCH; ++c) dst[(size_t)c * HWPIX] = v[c] * inv;
}

// ------------------------------------------- sigmoid(compat) prep (tiny) ----
// sigc: SIGROWS x SIGCOLS, zero padded outside the 21x21 live region.
__global__ __launch_bounds__(256) void sig_prep(const float* __restrict__ compat,
                                                float* __restrict__ sigc) {
  for (int idx = threadIdx.x; idx < SIGROWS * SIGCOLS; idx += 256) {
    int o = idx / SIGCOLS, i = idx % SIGCOLS;
    float val = 0.f;
    if (o < NCH && i < NCH) {
      float x = compat[o * NCH + i];
      val = __builtin_amdgcn_rcpf(1.0f + __expf(-x));
    }
    sigc[idx] = val;
  }
}

// ----------------------------------------------------------- CRF stencil ----
// 16x16 pixel tile / block; 22x22 halo of F (32ch) and Q (21ch) in LDS,
// pixel-major with padded strides so the hot loop uses ds_load_b128.
__global__ __launch_bounds__(256) void crf_pass(const float* __restrict__ F,
                                                const float* __restrict__ Qg,
                                                const float* __restrict__ wptr,
                                                float* __restrict__ out) {
  __shared__ float sF[HE * FSTR];   // 69,696 B
  __shared__ float sQ[HE * QSTR];   // 46,464 B  (total 113.4 KB < 320 KB/WGP)

  const int b  = blockIdx.z;
  const int x0 = blockIdx.x * TS, y0 = blockIdx.y * TS;
  const int tid = threadIdx.y * TS + threadIdx.x;

  // Stage F halo (zero-padded like jnp.pad). Plane-major loop -> coalesced.
  for (int idx = tid; idx < CCH * HE; idx += 256) {
    int c = idx / HE, e = idx % HE;
    int y = e / HT, x = e % HT;
    int gy = y0 + y - PADR, gx = x0 + x - PADR;
    float* dst = &sF[e * FSTR + c];
    bool inb = ((unsigned)gy < HH) && ((unsigned)gx < WW);
#ifdef USE_ASYNC_LDS
    if (inb) async_copy_b32(&F[(((size_t)b * CCH + c) * HH + gy) * WW + gx], dst);
    else     *dst = 0.f;
#else
    *dst = inb ? F[(((size_t)b * CCH + c) * HH + gy) * WW + gx] : 0.f;
#endif
  }
  // Stage Q halo; pad channels 21..23 with zeros.
  for (int idx = tid; idx < QSTR * HE; idx += 256) {
    int c = idx / HE, e = idx % HE;
    int y = e / HT, x = e % HT;
    int gy = y0 + y - PADR, gx = x0 + x - PADR;
    float* dst = &sQ[e * QSTR + c];
    bool inb = (c < NCH) && ((unsigned)gy < HH) && ((unsigned)gx < WW);
#ifdef USE_ASYNC_LDS
    if (inb) async_copy_b32(&Qg[(((size_t)b * NCH + c) * HH + gy) * WW + gx], dst);
    else     *dst = 0.f;
#else
    *dst = inb ? Qg[(((size_t)b * NCH + c) * HH + gy) * WW + gx] : 0.f;
#endif
  }
#ifdef USE_ASYNC_LDS
  async_wait_all();                 // s_wait_asynccnt 0 before the barrier
#endif
  __syncthreads();

  const float w  = wptr[0];
  const int tx = threadIdx.x, ty = threadIdx.y;
  const float4* sF4 = (const float4*)sF;
  const float4* sQ4 = (const float4*)sQ;

  // Center-pixel feature vector in registers (8 x float4 = 32 ch).
  const int ec = (ty + PADR) * HT + (tx + PADR);
  float4 fc[8];
#pragma unroll
  for (int k = 0; k < 8; ++k) fc[k] = sF4[ec * (FSTR / 4) + k];

  float acc[NCH];
#pragma unroll
  for (int o = 0; o < NCH; ++o) acc[o] = 0.f;

#pragma unroll 1
  for (int di = 0; di < KL; ++di) {
#pragma unroll
    for (int dj = 0; dj < KL; ++dj) {
      const int e = (ty + di) * HT + (tx + dj);
      float d2 = 0.f;
#pragma unroll
      for (int k = 0; k < 8; ++k) {
        float4 v = sF4[e * (FSTR / 4) + k];
        float a0 = fc[k].x - v.x, a1 = fc[k].y - v.y;
        float a2 = fc[k].z - v.z, a3 = fc[k].w - v.w;
        d2 += a0 * a0 + a1 * a1 + a2 * a2 + a3 * a3;
      }
      const float kern = w * __expf(-d2 * (1.0f / CCH));
#pragma unroll
      for (int k = 0; k < 5; ++k) {                 // 20 channels as float4
        float4 q = sQ4[e * (QSTR / 4) + k];
        acc[4 * k + 0] += kern * q.x;
        acc[4 * k + 1] += kern * q.y;
        acc[4 * k + 2] += kern * q.z;
        acc[4 * k + 3] += kern * q.w;
      }
      acc[20] += kern * sQ[e * QSTR + 20];          // channel 20
    }
  }

  const int gy = y0 + ty, gx = x0 + tx;
  const float inv_k = 1.0f / (float)KL;
#pragma unroll
  for (int o = 0; o < NCH; ++o)
    out[(((size_t)b * NCH + o) * HH + gy) * WW + gx] = acc[o] * inv_k;
}

// --------------------------------------------- WMMA compat-mix epilogue -----
// out = logit - (sigc @ crf + bias), as fp32 WMMA 16x16x4 (fp32-exact).
// Per wave: N = 16 pixels, two M tiles (o 0..15, 16..20), K = 21 padded to 24.
__global__ __launch_bounds__(256) void compat_epilogue(const float* __restrict__ logit,
                                                       const float* __restrict__ crf,
                                                       const float* __restrict__ sigc,
                                                       const float* __restrict__ bias,
                                                       float* __restrict__ out) {
  const int wave = threadIdx.x >> 5;      // 8 waves / block
  const int lane = threadIdx.x & 31;
  const int half = lane >> 4;             // 0: K={0,1}, D rows 0..7 ; 1: K={2,3}, rows 8..15
  const int n    = lane & 15;             // N column / A-matrix row
  const int pix  = (blockIdx.x * 8 + wave) * 16 + n;
  const int b = pix / HWPIX, p = pix % HWPIX;

  // A fragments per documented 16x4 f32 layout:
  //   lanes 0-15: M=lane, VGPR0=K0, VGPR1=K1 ; lanes 16-31: M=lane-16, K2/K3.
  const v2f* s2 = (const v2f*)sigc;
  v2f a0[6], a1[6];
#pragma unroll
  for (int kc = 0; kc < 6; ++kc) {
    int i0 = kc * 4 + half * 2;
    a0[kc] = s2[(n * SIGCOLS + i0) >> 1];
    a1[kc] = s2[((n + 16) * SIGCOLS + i0) >> 1];   // rows 21..31 are zero-padded
  }

  v8f acc0 = {0.f, 0.f, 0.f, 0.f, 0.f, 0.f, 0.f, 0.f};
  v8f acc1 = {0.f, 0.f, 0.f, 0.f, 0.f, 0.f, 0.f, 0.f};

#pragma unroll
  for (int kc = 0; kc < 6; ++kc) {
    int i0 = kc * 4 + half * 2;
    // B fragment (4x16, KxN): VGPR0 = rows K0|K2, VGPR1 = rows K1|K3.
    v2f bf;
    bf.x = (i0     < NCH) ? crf[((size_t)b * NCH + i0)     * HWPIX + p] : 0.f;
    bf.y = (i0 + 1 < NCH) ? crf[((size_t)b * NCH + i0 + 1) * HWPIX + p] : 0.f;
    acc0 = __builtin_amdgcn_wmma_f32_16x16x4_f32(false, a0[kc], false, bf,
                                                 (short)0, acc0, false, false);
    acc1 = __builtin_amdgcn_wmma_f32_16x16x4_f32(false, a1[kc], false, bf,
                                                 (short)0, acc1, false, false);
  }

  // D layout: VGPR v -> row v (lanes 0-15) / row v+8 (lanes 16-31), col n.
#pragma unroll
  for (int v = 0; v < 8; ++v) {
    int o = half * 8 + v;                      // 0..15
    size_t idx = ((size_t)b * NCH + o) * HWPIX + p;
    out[idx] = logit[idx] - (acc0[v] + bias[o]);
  }
  if (half == 0) {
#pragma unroll
    for (int v = 0; v < 5; ++v) {              // o = 16..20 from tile 1
      int o = 16 + v;
      size_t idx = ((size_t)b * NCH + o) * HWPIX + p;
      out[idx] = logit[idx] - (acc1[v] + bias[o]);
    }
  }
}

// ------------------------------------------------------------------ launch --
extern "C" void kernel_launch(void* const* d_in, const int* in_sizes, int n_in,
                              void* d_out, int out_size, void* d_ws, size_t ws_size,
                              hipStream_t stream) {
  (void)in_sizes; (void)n_in; (void)out_size; (void)ws_size;
  const float* F      = (const float*)d_in[0];
  const float* logit  = (const float*)d_in[1];
  const float* weight = (const float*)d_in[2];
  const float* compat = (const float*)d_in[3];
  const float* bias   = (const float*)d_in[4];
  // d_in[5] = kernel_len (==7), baked into the kernels.

  float* Q    = (float*)d_ws;                        // 34.4 MB
  float* crf  = Q + (size_t)BATCHN * NCH * HWPIX;    // 34.4 MB
  float* sigc = crf + (size_t)BATCHN * NCH * HWPIX;  // 3 KB padded sigmoid table
  float* out  = (float*)d_out;

  softmax_nc<<<NPIX / 256, 256, 0, stream>>>(logit, Q);
  sig_prep<<<1, 256, 0, stream>>>(compat, sigc);

  dim3 gB(WW / TS, HH / TS, BATCHN);
  crf_pass<<<gB, dim3(TS, TS), 0, stream>>>(F, Q, weight, crf);

  compat_epilogue<<<NPIX / 128, 256, 0, stream>>>(logit, crf, sigc, bias, out);
}